// GraphConv_86569360818435
// MI455X (gfx1250) — compile-verified
//
#include <hip/hip_runtime.h>
#include <hip/hip_bf16.h>

#define NN 50000
#define NE 800000
#define D  96
#define MTILES (NN / 16)   // 3125, exact

typedef __attribute__((ext_vector_type(2))) float v2f;
typedef __attribute__((ext_vector_type(8))) float v8f;

// ---------------- kernel 1: zero degrees + agg ----------------
__global__ void gc_zero(float* ws, int n) {
    int i = blockIdx.x * blockDim.x + threadIdx.x;
    for (; i < n; i += gridDim.x * blockDim.x) ws[i] = 0.0f;
}

// ---------------- kernel 2: degree counts ----------------
__global__ void gc_degrees(const int* __restrict__ src, const int* __restrict__ dst,
                           float* __restrict__ deg_out, float* __restrict__ deg_in) {
    int e = blockIdx.x * blockDim.x + threadIdx.x;
    if (e >= NE) return;
    atomicAdd(&deg_out[src[e]], 1.0f);
    atomicAdd(&deg_in[dst[e]], 1.0f);
}

// ---------------- kernel 3: edge gather/scatter-add ----------------
// 24 lanes per edge, each lane owns a float4 chunk of the 96-wide feature row.
__global__ void gc_edge_agg(const float* __restrict__ feat,
                            const int* __restrict__ src, const int* __restrict__ dst,
                            const float* __restrict__ deg_out, float* __restrict__ agg) {
    unsigned tid = blockIdx.x * blockDim.x + threadIdx.x;
    if (tid >= (unsigned)NE * 24u) return;
    int e = (int)(tid / 24u);
    int c = (int)(tid % 24u);           // chunk of 4 floats
    int s = src[e];
    int d = dst[e];
    float ds = deg_out[s];
    float ci = 1.0f / sqrtf(ds < 1.0f ? 1.0f : ds);
    const float4 v = *(const float4*)(feat + (size_t)s * D + c * 4);
    float* p = agg + (size_t)d * D + c * 4;
    atomicAdd(p + 0, v.x * ci);
    atomicAdd(p + 1, v.y * ci);
    atomicAdd(p + 2, v.z * ci);
    atomicAdd(p + 3, v.w * ci);
}

// ---------------- kernel 4: h = in_deg>0 ? agg*cj : feat (in place) ----------------
__global__ void gc_finalize_h(const float* __restrict__ feat,
                              const float* __restrict__ deg_in, float* __restrict__ hbuf) {
    unsigned tid = blockIdx.x * blockDim.x + threadIdx.x;
    if (tid >= (unsigned)NN * 24u) return;
    int n = (int)(tid / 24u);
    int c = (int)(tid % 24u);
    float di = deg_in[n];
    float4 r;
    if (di > 0.0f) {
        float cj = 1.0f / sqrtf(di);
        float4 a = *(const float4*)(hbuf + (size_t)n * D + c * 4);
        r.x = a.x * cj; r.y = a.y * cj; r.z = a.z * cj; r.w = a.w * cj;
    } else {
        r = *(const float4*)(feat + (size_t)n * D + c * 4);
    }
    *(float4*)(hbuf + (size_t)n * D + c * 4) = r;
}

// ---------------- kernel 5: out = h @ W + bias via V_WMMA_F32_16X16X4_F32 ----------------
// One wave per 16-row stripe of h; 6 accumulator tiles cover all 96 output cols.
// 32-bit A (16x4): lanes 0-15 hold K={0,1}, lanes 16-31 hold K={2,3}.
// 32-bit B (4x16): VGPR0 = rows {K, K+2} striped over lane halves; VGPR1 = rows {K+1, K+3}.
// 32-bit C (16x16): VGPR g -> row g (lanes 0-15) / row g+8 (lanes 16-31), col = lane&15.
__global__ __launch_bounds__(128) void gc_gemm_wmma(const float* __restrict__ h,
                                                    const float* __restrict__ W,
                                                    const float* __restrict__ bias,
                                                    float* __restrict__ out) {
    int wave  = threadIdx.x >> 5;
    int mtile = blockIdx.x * 4 + wave;
    if (mtile >= MTILES) return;          // wave-uniform guard: EXEC all-1s inside

    int lane = threadIdx.x & 31;
    int row  = lane & 15;
    int half = lane >> 4;                 // 0: K,K+1   1: K+2,K+3
    int m0   = mtile * 16;

    v8f acc[6];
#pragma unroll
    for (int t = 0; t < 6; ++t) acc[t] = (v8f){};

    const float* arow = h + (size_t)(m0 + row) * D;

#pragma unroll 4
    for (int k = 0; k < D; k += 4) {
        int kk = k + 2 * half;
        v2f a;
        float2 av = *(const float2*)(arow + kk);   // kk is even -> 8B aligned
        a.x = av.x; a.y = av.y;
#pragma unroll
        for (int t = 0; t < 6; ++t) {
            int col = t * 16 + row;
            v2f b;
            b.x = W[(size_t)kk * D + col];
            b.y = W[(size_t)(kk + 1) * D + col];
            acc[t] = __builtin_amdgcn_wmma_f32_16x16x4_f32(
                false, a, false, b, (short)0, acc[t], false, false);
        }
    }

#pragma unroll
    for (int t = 0; t < 6; ++t) {
        int col = t * 16 + row;
        float bc = bias[col];
#pragma unroll
        for (int g = 0; g < 8; ++g) {
            int m = m0 + g + 8 * half;
            out[(size_t)m * D + col] = acc[t][g] + bc;
        }
    }
}

extern "C" void kernel_launch(void* const* d_in, const int* in_sizes, int n_in,
                              void* d_out, int out_size, void* d_ws, size_t ws_size,
                              hipStream_t stream) {
    const float* feat   = (const float*)d_in[0];   // [NN, D]
    const float* weight = (const float*)d_in[1];   // [D, D]
    const float* bias   = (const float*)d_in[2];   // [D]
    const int*   src    = (const int*)d_in[3];     // [NE]
    const int*   dst    = (const int*)d_in[4];     // [NE]
    float* out = (float*)d_out;

    float* ws      = (float*)d_ws;
    float* deg_out = ws;                 // NN floats
    float* deg_in  = ws + NN;            // NN floats
    float* hbuf    = ws + 2 * NN;        // NN*D floats (agg, then h in place)

    // 1) zero degrees + agg
    int nzero = 2 * NN + NN * D;
    gc_zero<<<(nzero + 255) / 256, 256, 0, stream>>>(ws, nzero);

    // 2) degrees
    gc_degrees<<<(NE + 255) / 256, 256, 0, stream>>>(src, dst, deg_out, deg_in);

    // 3) edge scatter-add
    unsigned eth = (unsigned)NE * 24u;
    gc_edge_agg<<<(eth + 191u) / 192u, 192, 0, stream>>>(feat, src, dst, deg_out, hbuf);

    // 4) finalize h
    unsigned nth = (unsigned)NN * 24u;
    gc_finalize_h<<<(nth + 191u) / 192u, 192, 0, stream>>>(feat, deg_in, hbuf);

    // 5) WMMA GEMM
    gc_gemm_wmma<<<(MTILES + 3) / 4, 128, 0, stream>>>(hbuf, weight, bias, out);
}